// PatchedQwen2AttentionMuxi_35562329211856
// MI455X (gfx1250) — compile-verified
//
#include <hip/hip_runtime.h>
#include <hip/hip_bf16.h>

typedef __attribute__((ext_vector_type(2))) float v2f;
typedef __attribute__((ext_vector_type(8))) float v8f;

constexpr int BATCH = 64;
constexpr int HID   = 3584;
constexpr int NH    = 28;
constexpr int NKV   = 4;
constexpr int GQ    = 7;      // NH / NKV
constexpr int HD    = 128;
constexpr int PBS   = 64;     // page block size
constexpr int SEQ   = 2048;
constexpr int NSPL  = 8;      // flash-decoding splits
constexpr int CHK   = SEQ / NSPL; // 256 keys per split

// ---------------------------------------------------------------------------
// GEMM async stage: copy one 32-K chunk of W (32x16) and this wave's A tile
// (16x32) into LDS with async b128 copies. 5 async ops per thread.
// ---------------------------------------------------------------------------
__device__ __forceinline__ void gemm_issue_chunk(
    const float* __restrict__ A, const float* __restrict__ W,
    int K, int N, int m0, int n0, int k0,
    unsigned ldw_base, unsigned lda_base, int tid, int lane) {
  { // W[k0..k0+31][n0..n0+15] row-major into LDS, 1 float4 per thread
    const int krow = tid >> 2;
    const int wcol = (tid & 3) * 4;
    const unsigned ldst = ldw_base + (unsigned)((krow * 16 + wcol) * 4);
    const unsigned long long gsrc =
        (unsigned long long)(uintptr_t)(W + (size_t)(k0 + krow) * N + n0 + wcol);
    asm volatile("global_load_async_to_lds_b128 %0, %1, off"
                 :: "v"(ldst), "v"(gsrc) : "memory");
  }
#pragma unroll
  for (int u = 0; u < 4; ++u) { // per-wave A rows m0..m0+15, k0..k0+31
    const int i   = lane + u * 32;   // float4 index 0..127
    const int row = i >> 3, seg = i & 7;
    const unsigned ldst = lda_base + (unsigned)((row * 32 + seg * 4) * 4);
    const unsigned long long gsrc =
        (unsigned long long)(uintptr_t)(A + (size_t)(m0 + row) * K + k0 + seg * 4);
    asm volatile("global_load_async_to_lds_b128 %0, %1, off"
                 :: "v"(ldst), "v"(gsrc) : "memory");
  }
}

// ---------------------------------------------------------------------------
// Skinny GEMM: out(64 x N) = A(64 x K) @ W(K x N) (+ bias). 4 waves / block,
// wave w -> rows 16w, block -> 16 columns. Double-buffered async LDS staging:
// chunk n+1's copies fly while chunk n feeds the WMMA stream from DS.
// ---------------------------------------------------------------------------
__global__ __launch_bounds__(128)
void gemm_m64_wmma(const float* __restrict__ A, const float* __restrict__ W,
                   const float* __restrict__ bias, float* __restrict__ out,
                   int K, int N) {
  __shared__ float lds_w[2][32 * 16];     // [k][n] W chunk
  __shared__ float lds_a[2][4][16 * 32];  // per-wave [row][k] A chunk
  const int tid  = threadIdx.x;
  const int wave = tid >> 5;
  const int lane = tid & 31;
  const int hv   = lane >> 4;
  const int l16  = lane & 15;
  const int m0   = wave * 16;
  const int n0   = blockIdx.x * 16;
  const int nchunk = K >> 5;

  const unsigned lw0 = (unsigned)(uintptr_t)&lds_w[0][0];
  const unsigned lw1 = (unsigned)(uintptr_t)&lds_w[1][0];
  const unsigned la0 = (unsigned)(uintptr_t)&lds_a[0][wave][0];
  const unsigned la1 = (unsigned)(uintptr_t)&lds_a[1][wave][0];

  v8f c = {};
  gemm_issue_chunk(A, W, K, N, m0, n0, 0, lw0, la0, tid, lane);
  for (int ck = 0; ck < nchunk; ++ck) {
    const int  cur      = ck & 1;
    const bool has_next = (ck + 1 < nchunk);
    if (has_next)
      gemm_issue_chunk(A, W, K, N, m0, n0, (ck + 1) * 32,
                       cur ? lw0 : lw1, cur ? la0 : la1, tid, lane);
    // oldest 5 copies (current chunk) complete; next chunk's 5 may stay in flight
    if (has_next) asm volatile("s_wait_asynccnt 0x5" ::: "memory");
    else          asm volatile("s_wait_asynccnt 0x0" ::: "memory");
    __syncthreads();
#pragma unroll
    for (int kk = 0; kk < 8; ++kk) {
      v2f a = *(const v2f*)(&lds_a[cur][wave][l16 * 32 + kk * 4 + 2 * hv]);
      v2f bb;
      bb.x = lds_w[cur][(kk * 4 + 2 * hv + 0) * 16 + l16];
      bb.y = lds_w[cur][(kk * 4 + 2 * hv + 1) * 16 + l16];
      c = __builtin_amdgcn_wmma_f32_16x16x4_f32(false, a, false, bb,
                                                (short)0, c, false, false);
    }
    __syncthreads();  // all waves done reading before buffer is overwritten
  }
  const float bv = bias ? bias[n0 + l16] : 0.0f;
#pragma unroll
  for (int i = 0; i < 8; ++i)
    out[(size_t)(m0 + i + 8 * hv) * N + n0 + l16] = c[i] + bv;
}

// ---------------------------------------------------------------------------
// RoPE on q (B,NH,HD) and k (B,NKV,HD), in workspace buffers.
// ---------------------------------------------------------------------------
__global__ __launch_bounds__(64)
void rope_kernel(float* __restrict__ q, float* __restrict__ k,
                 const int* __restrict__ pos_ids) {
  const int b = blockIdx.x;
  const int j = threadIdx.x;                 // 0..63 = half of HD
  const float pos = (float)pos_ids[b];
  const float inv = __powf(1.0e6f, -(float)j / 64.0f);
  float s, c;
  __sincosf(pos * inv, &s, &c);
  for (int h = 0; h < NH; ++h) {
    float* p = q + ((size_t)b * NH + h) * HD;
    const float x1 = p[j], x2 = p[j + 64];
    p[j]      = x1 * c - x2 * s;
    p[j + 64] = x2 * c + x1 * s;
  }
  for (int h = 0; h < NKV; ++h) {
    float* p = k + ((size_t)b * NKV + h) * HD;
    const float x1 = p[j], x2 = p[j + 64];
    p[j]      = x1 * c - x2 * s;
    p[j + 64] = x2 * c + x1 * s;
  }
}

// ---------------------------------------------------------------------------
// Issue one 16-key K/V tile (2 x 8KB) as async global->LDS b128 copies.
// 32 async instructions per wave -> ASYNCcnt += 32.
// ---------------------------------------------------------------------------
__device__ __forceinline__ void issue_tile_async(
    const float* __restrict__ k_cache, const float* __restrict__ v_cache,
    const int* __restrict__ block_offsets, int b, int kvh, int t0,
    unsigned lkbase, unsigned lvbase, int tid) {
  const int    blk     = block_offsets[b * (SEQ / PBS) + (t0 / PBS)];
  const size_t rowbase = (size_t)blk * PBS + (t0 % PBS);
  for (int i = tid; i < 16 * 32; i += 32) {
    const int      key  = i >> 5, seg = i & 31;
    const size_t   gofs = ((rowbase + key) * NKV + kvh) * HD + seg * 4;
    const unsigned loff = (unsigned)((key * HD + seg * 4) * 4);
    const unsigned lk   = lkbase + loff;
    const unsigned lv   = lvbase + loff;
    const unsigned long long gk = (unsigned long long)(uintptr_t)(k_cache + gofs);
    const unsigned long long gv = (unsigned long long)(uintptr_t)(v_cache + gofs);
    asm volatile("global_load_async_to_lds_b128 %0, %1, off"
                 :: "v"(lk), "v"(gk) : "memory");
    asm volatile("global_load_async_to_lds_b128 %0, %1, off"
                 :: "v"(lv), "v"(gv) : "memory");
  }
}

// ---------------------------------------------------------------------------
// Flash-decoding partial attention. One wave32 workgroup per (b, kv, split).
// Double-buffered async K/V tiles in LDS; WMMA f32 16x16x4 for QK^T and PV;
// online softmax in the C-matrix lane layout; Q held in registers.
// ---------------------------------------------------------------------------
__global__ __launch_bounds__(32)
void attn_partial(const float* __restrict__ qbuf,
                  const float* __restrict__ knew,
                  const float* __restrict__ vnew,
                  const float* __restrict__ k_cache,
                  const float* __restrict__ v_cache,
                  const int* __restrict__ block_offsets,
                  const int* __restrict__ kv_seq_length,
                  float* __restrict__ o_part,   // [B*NKV*NSPL][8][HD]
                  float* __restrict__ m_part,   // [B*NKV*NSPL][8]
                  float* __restrict__ l_part) { // [B*NKV*NSPL][8]
  __shared__ float lds_k[2][16 * HD];
  __shared__ float lds_v[2][16 * HD];
  __shared__ float lds_p[16 * 16];

  const int wg    = blockIdx.x;
  const int split = wg % NSPL;
  const int kv    = (wg / NSPL) % NKV;
  const int b     = wg / (NSPL * NKV);
  const int tid   = threadIdx.x;
  const int hv    = tid >> 4;
  const int l16   = tid & 15;

  const int   kvlen = kv_seq_length[b];
  const int   slot  = kvlen - 1;
  const float scale = 0.08838834764831845f;  // 1/sqrt(128)

  const unsigned lk0 = (unsigned)(uintptr_t)&lds_k[0][0];
  const unsigned lk1 = (unsigned)(uintptr_t)&lds_k[1][0];
  const unsigned lv0 = (unsigned)(uintptr_t)&lds_v[0][0];
  const unsigned lv1 = (unsigned)(uintptr_t)&lds_v[1][0];

  // Q A-operands in registers; rows >= GQ are zero padding (also guards OOB).
  v2f qa[32];
#pragma unroll
  for (int kk = 0; kk < 32; ++kk) {
    v2f q = {};
    if (l16 < GQ)
      q = *(const v2f*)(qbuf + ((size_t)b * NH + kv * GQ + l16) * HD + kk * 4 + 2 * hv);
    qa[kk] = q;
  }

  float mrow[8], lrow[8];
#pragma unroll
  for (int i = 0; i < 8; ++i) { mrow[i] = -1e30f; lrow[i] = 0.0f; }
  v8f oacc[8] = {};

  const int t_base = split * CHK;
  int nt = (kvlen - t_base + 15) >> 4;      // valid 16-key tiles in this split
  if (nt < 0) nt = 0;
  if (nt > CHK / 16) nt = CHK / 16;

  if (nt > 0)
    issue_tile_async(k_cache, v_cache, block_offsets, b, kv, t_base, lk0, lv0, tid);

  for (int tt = 0; tt < nt; ++tt) {
    const int  cur      = tt & 1;
    const int  t0       = t_base + tt * 16;
    const bool has_next = (tt + 1 < nt);
    if (has_next)
      issue_tile_async(k_cache, v_cache, block_offsets, b, kv, t0 + 16,
                       cur ? lk0 : lk1, cur ? lv0 : lv1, tid);
    // Wait for the current buffer's 32 copies (next tile's 32 may stay in flight).
    if (has_next) asm volatile("s_wait_asynccnt 0x20" ::: "memory");
    else          asm volatile("s_wait_asynccnt 0x0"  ::: "memory");
    __syncthreads();

    // Functional cache update: position kv_len-1 uses freshly projected k/v.
    if (slot >= t0 && slot < t0 + 16) {
      const int    row  = slot - t0;
      const size_t nofs = ((size_t)b * NKV + kv) * HD + tid * 4;
      *(float4*)(&lds_k[cur][row * HD + tid * 4]) = *(const float4*)(knew + nofs);
      *(float4*)(&lds_v[cur][row * HD + tid * 4]) = *(const float4*)(vnew + nofs);
    }
    __syncthreads();

    // scores(16 heads x 16 keys) = Q(16x128) @ K^T, 32 WMMA steps of K=4.
    const float* ldsk = &lds_k[cur][0];
    v8f sc = {};
#pragma unroll
    for (int kk = 0; kk < 32; ++kk) {
      v2f bb = *(const v2f*)(&ldsk[l16 * HD + kk * 4 + 2 * hv]);
      sc = __builtin_amdgcn_wmma_f32_16x16x4_f32(false, qa[kk], false, bb,
                                                 (short)0, sc, false, false);
    }

    // Online softmax per row (rows i / i+8 live in lane halves; width-16 xor).
    const int  t     = t0 + l16;
    const bool valid = (t < kvlen);
    float p[8], alpha[8];
#pragma unroll
    for (int i = 0; i < 8; ++i) {
      const float s  = sc[i] * scale;
      float       sv = valid ? s : -1e30f;
#pragma unroll
      for (int d = 1; d < 16; d <<= 1) sv = fmaxf(sv, __shfl_xor(sv, d, 16));
      const float mnew = fmaxf(mrow[i], sv);
      alpha[i] = __expf(mrow[i] - mnew);
      const float pe = valid ? __expf(s - mnew) : 0.0f;
      float psum = pe;
#pragma unroll
      for (int d = 1; d < 16; d <<= 1) psum += __shfl_xor(psum, d, 16);
      lrow[i] = lrow[i] * alpha[i] + psum;
      mrow[i] = mnew;
      p[i]    = pe;
    }
#pragma unroll
    for (int i = 0; i < 8; ++i)
      lds_p[(i + 8 * hv) * 16 + l16] = p[i];   // C layout -> LDS row-major
#pragma unroll
    for (int j = 0; j < 8; ++j)
#pragma unroll
      for (int i = 0; i < 8; ++i) oacc[j][i] *= alpha[i];
    __syncthreads();

    // O(16x128) += P(16x16) @ V(16x128): 8 col-tiles x 4 WMMA (K=4 keys).
    const float* ldsv = &lds_v[cur][0];
#pragma unroll
    for (int j = 0; j < 8; ++j) {
      v8f o = oacc[j];
#pragma unroll
      for (int kk = 0; kk < 4; ++kk) {
        v2f a = *(const v2f*)(&lds_p[l16 * 16 + kk * 4 + 2 * hv]);
        v2f bb;
        bb.x = ldsv[(kk * 4 + 2 * hv + 0) * HD + j * 16 + l16];
        bb.y = ldsv[(kk * 4 + 2 * hv + 1) * HD + j * 16 + l16];
        o = __builtin_amdgcn_wmma_f32_16x16x4_f32(false, a, false, bb,
                                                  (short)0, o, false, false);
      }
      oacc[j] = o;
    }
    __syncthreads();
  }

  // Emit un-normalized partials (numerator, running max, running sum).
  const size_t pbase = (size_t)((b * NKV + kv) * NSPL + split);
#pragma unroll
  for (int j = 0; j < 8; ++j)
#pragma unroll
    for (int i = 0; i < 8; ++i) {
      const int row = i + 8 * hv;
      if (row < GQ)
        o_part[(pbase * 8 + row) * HD + j * 16 + l16] = oacc[j][i];
    }
  if (tid == 0) {
#pragma unroll
    for (int i = 0; i < GQ; ++i) {
      m_part[pbase * 8 + i] = mrow[i];
      l_part[pbase * 8 + i] = lrow[i];
    }
  }
}

// ---------------------------------------------------------------------------
// Combine split partials with log-sum-exp weights.
// ---------------------------------------------------------------------------
__global__ __launch_bounds__(128)
void attn_combine(const float* __restrict__ o_part,
                  const float* __restrict__ m_part,
                  const float* __restrict__ l_part,
                  float* __restrict__ attn) {
  const int bkv = blockIdx.x;
  const int b = bkv / NKV, kv = bkv % NKV;
  const int d = threadIdx.x;
  for (int row = 0; row < GQ; ++row) {
    float M = -1e30f;
    for (int s = 0; s < NSPL; ++s)
      M = fmaxf(M, m_part[((size_t)bkv * NSPL + s) * 8 + row]);
    float L = 0.0f, acc = 0.0f;
    for (int s = 0; s < NSPL; ++s) {
      const size_t ps = (size_t)bkv * NSPL + s;
      const float  w  = __expf(m_part[ps * 8 + row] - M);
      L   += w * l_part[ps * 8 + row];
      acc += w * o_part[(ps * 8 + row) * HD + d];
    }
    attn[((size_t)b * NH + kv * GQ + row) * HD + d] = acc / L;
  }
}

// ---------------------------------------------------------------------------
extern "C" void kernel_launch(void* const* d_in, const int* in_sizes, int n_in,
                              void* d_out, int out_size, void* d_ws, size_t ws_size,
                              hipStream_t stream) {
  const float* hidden        = (const float*)d_in[0];
  const float* q_w           = (const float*)d_in[1];
  const float* q_b           = (const float*)d_in[2];
  const float* k_w           = (const float*)d_in[3];
  const float* k_b           = (const float*)d_in[4];
  const float* v_w           = (const float*)d_in[5];
  const float* v_b           = (const float*)d_in[6];
  const float* o_w           = (const float*)d_in[7];
  const float* k_cache       = (const float*)d_in[8];
  const float* v_cache       = (const float*)d_in[9];
  const int*   block_offsets = (const int*)d_in[10];
  const int*   kv_len        = (const int*)d_in[11];
  const int*   pos_ids       = (const int*)d_in[12];
  float*       out           = (float*)d_out;

  float* q_buf    = (float*)d_ws;                       // B*NH*HD
  float* k_buf    = q_buf    + (size_t)BATCH * NH * HD; // B*NKV*HD
  float* v_buf    = k_buf    + (size_t)BATCH * NKV * HD;
  float* attn_buf = v_buf    + (size_t)BATCH * NKV * HD;
  float* o_part   = attn_buf + (size_t)BATCH * NH * HD; // B*NKV*NSPL*8*HD
  float* m_part   = o_part   + (size_t)BATCH * NKV * NSPL * 8 * HD;
  float* l_part   = m_part   + (size_t)BATCH * NKV * NSPL * 8;

  // QKV projections (WMMA GEMMs, M=64, async double-buffered staging)
  gemm_m64_wmma<<<(NH * HD) / 16, 128, 0, stream>>>(hidden, q_w, q_b, q_buf, HID, NH * HD);
  gemm_m64_wmma<<<(NKV * HD) / 16, 128, 0, stream>>>(hidden, k_w, k_b, k_buf, HID, NKV * HD);
  gemm_m64_wmma<<<(NKV * HD) / 16, 128, 0, stream>>>(hidden, v_w, v_b, v_buf, HID, NKV * HD);
  // RoPE on q and k
  rope_kernel<<<BATCH, 64, 0, stream>>>(q_buf, k_buf, pos_ids);
  // Flash-decoding attention (split-S, async double-buffered K/V) + combine
  attn_partial<<<BATCH * NKV * NSPL, 32, 0, stream>>>(q_buf, k_buf, v_buf,
                                                      k_cache, v_cache,
                                                      block_offsets, kv_len,
                                                      o_part, m_part, l_part);
  attn_combine<<<BATCH * NKV, 128, 0, stream>>>(o_part, m_part, l_part, attn_buf);
  // Output projection
  gemm_m64_wmma<<<HID / 16, 128, 0, stream>>>(attn_buf, o_w, nullptr, out, NH * HD, HID);
}